// MultiHeadAttention_29394756173884
// MI455X (gfx1250) — compile-verified
//
#include <hip/hip_runtime.h>
#include <math.h>

#define CC   512
#define TT   2048
#define HH   8
#define DD   64
#define WIN  10
#define BAND 256

typedef __attribute__((ext_vector_type(2))) float v2f;
typedef __attribute__((ext_vector_type(8))) float v8f;

__device__ __forceinline__ v8f wmma_f32_16x16x4(v2f a, v2f b, v8f c) {
  // D = A(16x4) * B(4x16) + C(16x16), fp32, wave32
  return __builtin_amdgcn_wmma_f32_16x16x4_f32(false, a, false, b, (short)0, c,
                                               false, false);
}

// ---------------------------------------------------------------------------
// Kernel 1: fused QKV projections.  {q,k,v}[o,t] = W[o,:] . X[:,t] + b[o]
// One wave computes one 16x16 output tile; K=512 swept in steps of 4.
// ---------------------------------------------------------------------------
__global__ void __launch_bounds__(256) qkv_proj_kernel(
    const float* __restrict__ x, const float* __restrict__ cin,
    const float* __restrict__ wq, const float* __restrict__ bq,
    const float* __restrict__ wk, const float* __restrict__ bk,
    const float* __restrict__ wv, const float* __restrict__ bv,
    float* __restrict__ q, float* __restrict__ k, float* __restrict__ v) {
  int wid  = blockIdx.x * (blockDim.x >> 5) + (threadIdx.x >> 5);
  int lane = threadIdx.x & 31;
  const int NT = TT / 16;   // 128 col tiles
  const int MT = CC / 16;   // 32 row tiles
  int which = wid / (MT * NT);
  int rem   = wid % (MT * NT);
  int mt = rem / NT, nt = rem % NT;
  const float* Wm   = (which == 0) ? wq : (which == 1) ? wk : wv;
  const float* bias = (which == 0) ? bq : (which == 1) ? bk : bv;
  const float* X    = (which == 0) ? x  : cin;
  float*       Dst  = (which == 0) ? q  : (which == 1) ? k : v;

  int hlf = lane >> 4;      // 0: K lanes 0..1, 1: K lanes 2..3 (ISA A/B striping)
  int lid = lane & 15;
  int arow = mt * 16 + lid; // A: M = lid
  int bcol = nt * 16 + lid; // B: N = lid
  v8f acc = {};
  for (int kk = 0; kk < CC; kk += 4) {
    int ka = kk + hlf * 2;
    v2f a, b;
    a.x = Wm[arow * CC + ka];
    a.y = Wm[arow * CC + ka + 1];
    b.x = X[ka * TT + bcol];
    b.y = X[(ka + 1) * TT + bcol];
    acc = wmma_f32_16x16x4(a, b, acc);
  }
#pragma unroll
  for (int rr = 0; rr < 8; ++rr) {
    int row = mt * 16 + rr + hlf * 8;   // C layout: M = rr (+8 for upper half)
    Dst[row * TT + bcol] = acc[rr] + bias[row];
  }
}

// ---------------------------------------------------------------------------
// Kernel 2: relative-key bias table.
// relk[h,t,dl] = scale * sum_d q[h*64+d, t] * emb_k[dl, d],  dl in [0,21)
// (equals _rel_to_abs(rel_logits)[t, s] at s = t + dl - 10; zero elsewhere)
// ---------------------------------------------------------------------------
__global__ void __launch_bounds__(256) relk_kernel(
    const float* __restrict__ q, const float* __restrict__ emb_k,
    float* __restrict__ relk) {
  int idx = blockIdx.x * blockDim.x + threadIdx.x;
  if (idx >= HH * TT * 21) return;
  int dl = idx % 21;
  int t  = (idx / 21) % TT;
  int h  = idx / (21 * TT);
  float s = 0.f;
  for (int d = 0; d < DD; ++d)
    s += q[(h * DD + d) * TT + t] * emb_k[dl * DD + d];
  relk[idx] = s * 0.125f;   // scale = 1/sqrt(64), exact power of two
}

// ---------------------------------------------------------------------------
// Kernel 3: banded flash attention (+ proximal bias, mask, rel-k, rel-v).
// One wave handles one head-row-tile (16 query rows); streams 33 key tiles.
// ---------------------------------------------------------------------------
__global__ void __launch_bounds__(256) attn_kernel(
    const float* __restrict__ q, const float* __restrict__ kmat,
    const float* __restrict__ vmat, const int* __restrict__ mask,
    const float* __restrict__ relk, const float* __restrict__ emb_v,
    float* __restrict__ o_ct) {
  __shared__ float embv_s[21 * DD];
  __shared__ float p_s[8][16][17];     // P tile, C-layout in, A-layout out
  __shared__ float band_s[8][16][21];  // diagonal scores -> probabilities
  __shared__ float m_s[8][16];
  __shared__ float l_s[8][16];

  int tid = threadIdx.x;
  for (int i = tid; i < 21 * DD; i += 256) embv_s[i] = emb_v[i];
  __syncthreads();

  int wv   = tid >> 5;
  int lane = tid & 31;
  int hlf = lane >> 4, lid = lane & 15;
  int gw = blockIdx.x * 8 + wv;
  int h  = gw >> 7;          // / 128 row tiles
  int it = gw & 127;
  int t0 = it * 16;

  for (int i = lane; i < 16 * 21; i += 32)
    band_s[wv][i / 21][i % 21] = -INFINITY;   // unwritten (t,dl) -> p = 0

  // Preload Q A-fragments (K = 64 -> 16 wmma k-steps), reused for all 33 tiles
  v2f qa[16];
  int trow = t0 + lid;
#pragma unroll
  for (int ks = 0; ks < 16; ++ks) {
    int d0 = ks * 4 + hlf * 2;
    qa[ks].x = q[(h * DD + d0) * TT + trow];
    qa[ks].y = q[(h * DD + d0 + 1) * TT + trow];
  }

  float m[8], l[8];
  v8f oacc[4] = {};
#pragma unroll
  for (int rr = 0; rr < 8; ++rr) { m[rr] = -1e30f; l[rr] = 0.f; }

  for (int j = it - 16; j <= it + 16; ++j) {   // band: |16(i-j)| - 15 <= 256
    if (j < 0 || j >= TT / 16) continue;
    int s0 = j * 16;

    // S = Q_tile^T . K_tile  (scores before scale)
    v8f S = {};
#pragma unroll
    for (int ks = 0; ks < 16; ++ks) {
      int d0 = ks * 4 + hlf * 2;
      v2f b;
      b.x = kmat[(h * DD + d0) * TT + s0 + lid];
      b.y = kmat[(h * DD + d0 + 1) * TT + s0 + lid];
      S = wmma_f32_16x16x4(qa[ks], b, S);
    }

    int scol = s0 + lid;
    float sv[8];
#pragma unroll
    for (int rr = 0; rr < 8; ++rr) {
      int t   = t0 + rr + hlf * 8;
      float val = S[rr] * 0.125f;
      int dlt = scol - t;
      int ad  = dlt < 0 ? -dlt : dlt;
      if (ad <= WIN) val += relk[(h * TT + t) * 21 + (dlt + WIN)];
      val -= log1pf((float)ad);                 // proximal bias
      if (mask[t * TT + scol] == 0) val = -1e4f;
      if (ad > BAND) val = -1e4f;
      if (ad <= WIN) band_s[wv][rr + hlf * 8][dlt + WIN] = val;
      sv[rr] = val;
    }

    // Online softmax: 16-lane row reductions (xor masks stay within halves)
    float corr[8];
#pragma unroll
    for (int rr = 0; rr < 8; ++rr) {
      float rmax = sv[rr];
      for (int off = 1; off < 16; off <<= 1)
        rmax = fmaxf(rmax, __shfl_xor(rmax, off, 32));
      float mnew = fmaxf(m[rr], rmax);
      corr[rr]   = expf(m[rr] - mnew);
      float p    = expf(sv[rr] - mnew);
      float rsum = p;
      for (int off = 1; off < 16; off <<= 1)
        rsum += __shfl_xor(rsum, off, 32);
      l[rr] = l[rr] * corr[rr] + rsum;
      m[rr] = mnew;
      p_s[wv][rr + hlf * 8][lid] = p;   // C-layout -> LDS
    }
#pragma unroll
    for (int n = 0; n < 4; ++n)
#pragma unroll
      for (int rr = 0; rr < 8; ++rr) oacc[n][rr] = oacc[n][rr] * corr[rr];

    // oacc += P (from LDS, A-layout) x V_tile
#pragma unroll
    for (int n = 0; n < 4; ++n) {
      v8f accn = oacc[n];
#pragma unroll
      for (int ks = 0; ks < 4; ++ks) {
        int k0 = ks * 4 + hlf * 2;
        v2f a, b;
        a.x = p_s[wv][lid][k0];
        a.y = p_s[wv][lid][k0 + 1];
        int dg = h * DD + n * 16 + lid;
        b.x = vmat[dg * TT + s0 + k0];
        b.y = vmat[dg * TT + s0 + k0 + 1];
        accn = wmma_f32_16x16x4(a, b, accn);
      }
      oacc[n] = accn;
    }
  }

  // Publish per-row m/l, turn banded scores into normalized probabilities
  if (lid == 0) {
#pragma unroll
    for (int rr = 0; rr < 8; ++rr) {
      m_s[wv][rr + hlf * 8] = m[rr];
      l_s[wv][rr + hlf * 8] = l[rr];
    }
  }
  for (int i = lane; i < 16 * 21; i += 32) {
    int tl = i / 21, dl = i % 21;
    float sc = band_s[wv][tl][dl];
    band_s[wv][tl][dl] = expf(sc - m_s[wv][tl]) / l_s[wv][tl];
  }

  // out[t,d] = oacc/l + sum_dl p[t,t+dl-10] * emb_v[dl,d]; store as [C,T]
#pragma unroll
  for (int n = 0; n < 4; ++n) {
#pragma unroll
    for (int rr = 0; rr < 8; ++rr) {
      int tl = rr + hlf * 8;
      float o = oacc[n][rr] / l[rr];
      for (int dl = 0; dl < 21; ++dl)
        o += band_s[wv][tl][dl] * embv_s[dl * DD + n * 16 + lid];
      o_ct[(h * DD + n * 16 + lid) * TT + (t0 + tl)] = o;
    }
  }
}

// ---------------------------------------------------------------------------
// Kernel 4: output projection. out[o,t] = wo[o,:] . o_ct[:,t] + bo[o]
// ---------------------------------------------------------------------------
__global__ void __launch_bounds__(256) out_proj_kernel(
    const float* __restrict__ octb, const float* __restrict__ wo,
    const float* __restrict__ bo, float* __restrict__ out) {
  int wid  = blockIdx.x * (blockDim.x >> 5) + (threadIdx.x >> 5);
  int lane = threadIdx.x & 31;
  const int NT = TT / 16;
  int mt = wid / NT, nt = wid % NT;
  int hlf = lane >> 4, lid = lane & 15;
  int arow = mt * 16 + lid, bcol = nt * 16 + lid;
  v8f acc = {};
  for (int kk = 0; kk < CC; kk += 4) {
    int ka = kk + hlf * 2;
    v2f a, b;
    a.x = wo[arow * CC + ka];
    a.y = wo[arow * CC + ka + 1];
    b.x = octb[ka * TT + bcol];
    b.y = octb[(ka + 1) * TT + bcol];
    acc = wmma_f32_16x16x4(a, b, acc);
  }
#pragma unroll
  for (int rr = 0; rr < 8; ++rr) {
    int row = mt * 16 + rr + hlf * 8;
    out[row * TT + bcol] = acc[rr] + bo[row];
  }
}

// ---------------------------------------------------------------------------
extern "C" void kernel_launch(void* const* d_in, const int* in_sizes, int n_in,
                              void* d_out, int out_size, void* d_ws,
                              size_t ws_size, hipStream_t stream) {
  (void)in_sizes; (void)n_in; (void)out_size; (void)ws_size;
  const float* x  = (const float*)d_in[0];
  const float* c  = (const float*)d_in[1];
  const int*   am = (const int*)d_in[2];
  const float* wq = (const float*)d_in[3];
  const float* bq = (const float*)d_in[4];
  const float* wk = (const float*)d_in[5];
  const float* bk = (const float*)d_in[6];
  const float* wv = (const float*)d_in[7];
  const float* bv = (const float*)d_in[8];
  const float* wo = (const float*)d_in[9];
  const float* bo = (const float*)d_in[10];
  const float* ek = (const float*)d_in[11];
  const float* ev = (const float*)d_in[12];

  float* q    = (float*)d_ws;                 // [512, 2048]
  float* kbuf = q    + (size_t)CC * TT;       // [512, 2048]
  float* vbuf = kbuf + (size_t)CC * TT;       // [512, 2048]
  float* octb = vbuf + (size_t)CC * TT;       // [512, 2048] attention out (C,T)
  float* relk = octb + (size_t)CC * TT;       // [8, 2048, 21]

  // 3 * 32 * 128 tiles, 8 waves (16x16 tiles) per 256-thread block
  qkv_proj_kernel<<<dim3(3 * (CC / 16) * (TT / 16) / 8), 256, 0, stream>>>(
      x, c, wq, bq, wk, bk, wv, bv, q, kbuf, vbuf);
  relk_kernel<<<dim3((HH * TT * 21 + 255) / 256), 256, 0, stream>>>(q, ek, relk);
  attn_kernel<<<dim3(HH * (TT / 16) / 8), 256, 0, stream>>>(
      q, kbuf, vbuf, am, relk, ev, octb);
  out_proj_kernel<<<dim3((CC / 16) * (TT / 16) / 8), 256, 0, stream>>>(
      octb, wo, bo, (float*)d_out);
}